// SubManifold_42202348650932
// MI455X (gfx1250) — compile-verified
//
#include <hip/hip_runtime.h>
#include <cstdint>

// Problem constants (match reference)
#define KW 12                  // window size
#define DG 48                  // grid side
#define NTOK (DG * DG)         // 2304 tokens
#define NWIN 100               // (48-12)/4+1 = 10 starts per dim -> 100 windows
#define KK (KW * KW)           // 144
#define BSZ 8
#define WIN_ELEMS (KK * KK)    // 20736 floats per window  (81 KB)
#define HALF_ELEMS (WIN_ELEMS / 2)  // 10368 floats (41,472 B) per half-window

#if __has_builtin(__builtin_amdgcn_tensor_load_to_lds) && \
    __has_builtin(__builtin_amdgcn_tensor_store_from_lds) && \
    __has_builtin(__builtin_amdgcn_s_wait_tensorcnt)
#define USE_TDM 1
#else
#define USE_TDM 0
#endif

typedef uint32_t u32;
typedef uint64_t u64;
typedef __attribute__((ext_vector_type(4))) u32 u32x4;
typedef __attribute__((ext_vector_type(8))) int i32x8;
typedef __attribute__((ext_vector_type(4))) int i32x4;

// One workgroup (1 wave32) per (window, half): TDM gathers the strided 4-D tile
//   tile (cb,ca,rb,ra) = (12,12,12,6), strides (1,48,2304,110592) elements
// into LDS packed exactly in output row-major order, then a 1-D TDM store
// writes the 10368 contiguous floats to d_out.
__global__ __launch_bounds__(32) void SubManifold_tdm_gather(
    const float* __restrict__ x, float* __restrict__ out) {
  const u32 blk  = blockIdx.x;      // 0..1599
  const u32 wid  = blk >> 1;        // b*100 + w
  const u32 half = blk & 1u;        // which 6 row-groups (ra) of the window
  const u32 b  = wid / NWIN;
  const u32 w  = wid - b * NWIN;
  const u32 wr = w / 10u;
  const u32 wc = w - wr * 10u;
  const u32 R0 = 192u * wr + 4u * wc;   // window start token (row idx == col idx)

#if USE_TDM
  // Tile-start element: b*N^2 + (R0 + half*6*48)*N + R0
  const u64 srcElem = (u64)b * NTOK * NTOK + (u64)(R0 + half * 288u) * NTOK + (u64)R0;
  const u64 srcAddr = (u64)(uintptr_t)x + srcElem * 4u;
  const u64 dstAddr = (u64)(uintptr_t)out +
                      ((u64)wid * WIN_ELEMS + (u64)half * HALF_ELEMS) * 4u;

  const u32 BIG = 0x40000000u;  // "no OOB" tensor dims (tile coords max out at 11)

  // VADDR4 operand: unused per ISA (NULL); probe-verified zero-filled call.
  const i32x8 zero8 = {0, 0, 0, 0, 0, 0, 0, 0};

  // ---------------- load descriptor: 4-D strided gather -> packed LDS --------
  u32x4 lg0; i32x8 lg1; i32x4 lg2, lg3;
  lg0[0] = 1u;                                   // count=1 (valid user D#)
  lg0[1] = 0u;                                   // lds_addr = 0 (dyn LDS base)
  lg0[2] = (u32)srcAddr;                         // global_addr[31:0]
  lg0[3] = ((u32)(srcAddr >> 32) & 0x01FFFFFFu)  // global_addr[56:32]
           | 0x80000000u;                        // type=2 ("image")
  lg1[0] = (int)(2u << 16);                      // data_size=2 (4 bytes)
  lg1[1] = (int)((BIG & 0xFFFFu) << 16);         // tensor_dim0[15:0]
  lg1[2] = (int)((BIG >> 16) | ((BIG & 0xFFFFu) << 16)); // dim0 hi | dim1 lo
  lg1[3] = (int)((BIG >> 16) | (12u << 16));     // dim1 hi | tile_dim0=12 (cb)
  lg1[4] = (int)(12u | (12u << 16));             // tile_dim1=12 (ca) | tile_dim2=12 (rb)
  lg1[5] = (int)48u;                             // tensor_dim0_stride = 48 elems
  lg1[6] = (int)(2304u << 16);                   // s0 hi=0 | tensor_dim1_stride lo=2304
  lg1[7] = 0;                                    // tensor_dim1_stride hi
  lg2[0] = (int)BIG;                             // tensor_dim2
  lg2[1] = (int)BIG;                             // tensor_dim3
  lg2[2] = (int)110592u;                         // tensor_dim2_stride = 48*2304
  lg2[3] = (int)(6u << 16);                      // s2 hi=0 | tile_dim3=6 (ra half)
  lg3[0] = 0; lg3[1] = 0; lg3[2] = 0; lg3[3] = 0;

  __builtin_amdgcn_tensor_load_to_lds(lg0, lg1, lg2, lg3, zero8, 0);
  __builtin_amdgcn_s_wait_tensorcnt(0);          // LDS tile resident

  // ---------------- store descriptor: 1-D contiguous LDS -> out --------------
  u32x4 sg0; i32x8 sg1; i32x4 sg2, sg3;
  sg0[0] = 1u;
  sg0[1] = 0u;                                   // lds_addr = 0
  sg0[2] = (u32)dstAddr;
  sg0[3] = ((u32)(dstAddr >> 32) & 0x01FFFFFFu) | 0x80000000u;
  sg1[0] = (int)(2u << 16);                      // data_size = 4 bytes
  sg1[1] = (int)((BIG & 0xFFFFu) << 16);         // tensor_dim0 lo
  sg1[2] = (int)((BIG >> 16) | ((BIG & 0xFFFFu) << 16)); // dim0 hi | dim1 lo
  sg1[3] = (int)((BIG >> 16) | ((u32)HALF_ELEMS << 16)); // dim1 hi | tile_dim0=10368
  sg1[4] = (int)(1u | (1u << 16));               // tile_dim1=1 | tile_dim2=1
  sg1[5] = (int)HALF_ELEMS;                      // tensor_dim0_stride (unused, sane)
  sg1[6] = 0; sg1[7] = 0;
  sg2[0] = (int)BIG; sg2[1] = (int)BIG; sg2[2] = 0; sg2[3] = 0;
  sg3[0] = 0; sg3[1] = 0; sg3[2] = 0; sg3[3] = 0;

  __builtin_amdgcn_tensor_store_from_lds(sg0, sg1, sg2, sg3, zero8, 0);
  __builtin_amdgcn_s_wait_tensorcnt(0);
#else
  // Fallback (also what the host pass parses): aligned float4 gather with the
  // identical launch geometry. 12-float runs start 16B-aligned (pitch 9216 B,
  // col offsets multiples of 4 elements), output rows 576 B.
  const int tid = (int)threadIdx.x;
  const float* __restrict__ xb = x + (u64)b * NTOK * NTOK;
  float* __restrict__ o = out + (u64)wid * WIN_ELEMS + (u64)half * HALF_ELEMS;
  for (int v = tid; v < HALF_ELEMS / 4; v += 32) {
    const int lin = v * 4;                       // offset within this half
    const int i = (int)half * (KK / 2) + lin / KK;
    const int j = lin % KK;                      // multiple of 4, stays in a run
    const int r = (int)R0 + (i / KW) * DG + (i % KW);
    const int c = (int)R0 + (j / KW) * DG + (j % KW);
    const float4 val = *(const float4*)(xb + (u64)r * NTOK + c);
    *(float4*)(o + lin) = val;
  }
#endif
}

extern "C" void kernel_launch(void* const* d_in, const int* in_sizes, int n_in,
                              void* d_out, int out_size, void* d_ws, size_t ws_size,
                              hipStream_t stream) {
  (void)in_sizes; (void)n_in; (void)out_size; (void)d_ws; (void)ws_size;
  const float* x = (const float*)d_in[0];
  float* out = (float*)d_out;
  // 8 batches * 100 windows * 2 halves = 1600 workgroups, one wave32 each.
  // 41,472 B dynamic LDS per workgroup (TDM staging buffer; unused by fallback).
  SubManifold_tdm_gather<<<dim3(BSZ * NWIN * 2), dim3(32),
                           HALF_ELEMS * sizeof(float), stream>>>(x, out);
}